// EGATConv_35914516529581
// MI455X (gfx1250) — compile-verified
//
#include <hip/hip_runtime.h>
#include <hip/hip_bf16.h>
#include <math.h>

typedef float v2f __attribute__((ext_vector_type(2)));
typedef float v8f __attribute__((ext_vector_type(8)));
typedef unsigned int u32x4 __attribute__((ext_vector_type(4)));
typedef int i32x4 __attribute__((ext_vector_type(4)));
typedef int i32x8 __attribute__((ext_vector_type(8)));

#define C_IN   128
#define C_OUT  64
#define HEADS  4

#if __has_builtin(__builtin_amdgcn_tensor_load_to_lds)
#define HAVE_TDM 1
#endif

// ---------------------------------------------------------------------------
// Kernel 0: init out = bias, amax = -inf, denom = 0
// ---------------------------------------------------------------------------
__global__ __launch_bounds__(256) void k_init(float* __restrict__ out,
                                              const float* __restrict__ bias,
                                              float* __restrict__ amax,
                                              float* __restrict__ denom,
                                              int n) {
    int idx = blockIdx.x * blockDim.x + threadIdx.x;
    int total = n * C_OUT;
    if (idx < total) out[idx] = bias[idx & (C_OUT - 1)];
    if (idx < n * HEADS) {
        amax[idx]  = -INFINITY;
        denom[idx] = 0.0f;
    }
}

// ---------------------------------------------------------------------------
// Kernel 1: h = x @ weight via V_WMMA_F32_16X16X4_F32.
// weight (128x64 f32 = 32 KB) is staged into LDS once per workgroup by the
// Tensor Data Mover (D# descriptor per ISA ch.8), drained with
// s_wait_tensorcnt; an asm escape makes the TDM's LDS write visible to the
// compiler so the ds_load reads of B are not folded away.
// A 16x4 f32 layout: lanes0-15 M=0..15 {K0,K1}, lanes16-31 M=0..15 {K2,K3}
// B 4x16  f32 layout: lanes0-15 N=0..15 rows{K0,K1}, lanes16-31 rows{K2,K3}
// ---------------------------------------------------------------------------
__global__ __launch_bounds__(256) void k_gemm(const float* __restrict__ x,
                                              const float* __restrict__ w,
                                              float* __restrict__ h,
                                              int n) {
    __shared__ float lds_w[C_IN * C_OUT];   // 32 KB of the 320 KB WGP LDS

    const int wave = threadIdx.x >> 5;
    const int lane = threadIdx.x & 31;

#ifdef HAVE_TDM
    if (wave == 0) {
        // LDS byte offset of lds_w: low 32 bits of its generic address
        unsigned int lds_off = (unsigned int)(size_t)&lds_w[0];
        // ---- D# group 0 (128b): count=1, lds_addr, global_addr=w, type=2
        unsigned long long ga = (unsigned long long)(size_t)w;
        u32x4 g0;
        g0[0] = 1u;                                   // count=1, is_restore=0
        g0[1] = lds_off;                              // lds_addr (bytes)
        g0[2] = (unsigned int)(ga & 0xFFFFFFFFull);   // global_addr[31:0]
        g0[3] = (unsigned int)((ga >> 32) & 0x01FFFFFFull) | 0x80000000u; // type=2
        // ---- D# group 1 (256b): 2-D tile 64x128 of 4-byte elements
        i32x8 g1;
        g1[0] = 0x00020000;          // wg_mask=0, data_size=2 (4 bytes)
        g1[1] = (C_OUT << 16);       // tensor_dim0[15:0] in bits [31:16]
        g1[2] = (C_IN << 16);        // tensor_dim0[31:16]=0 | tensor_dim1[15:0]
        g1[3] = (C_OUT << 16);       // tensor_dim1[31:16]=0 | tile_dim0=64
        g1[4] = C_IN;                // tile_dim1=128, tile_dim2=0
        g1[5] = C_OUT;               // tensor_dim0_stride[31:0] = 64
        g1[6] = 0;
        g1[7] = 0;
        i32x4 z4 = {0, 0, 0, 0};
#if __clang_major__ >= 23
        i32x8 z8 = {0, 0, 0, 0, 0, 0, 0, 0};
        __builtin_amdgcn_tensor_load_to_lds(g0, g1, z4, z4, z8, 0);
#else
        __builtin_amdgcn_tensor_load_to_lds(g0, g1, z4, z4, 0);
#endif
        __builtin_amdgcn_s_wait_tensorcnt(0);
    }
#else
    for (int i = threadIdx.x; i < C_IN * C_OUT; i += 256) lds_w[i] = w[i];
#endif
    __syncthreads();
    // Make the TDM's LDS write visible to the compiler's alias analysis so
    // the B-matrix ds_loads below are not folded to undef.
    asm volatile("" : : "r"(&lds_w[0]) : "memory");

    const int m0 = (blockIdx.x * 8 + wave) * 16;
    if (m0 < n) {
        const int half = lane >> 4;          // 0: lanes 0-15, 1: lanes 16-31
        const int l16  = lane & 15;
        int rowA = m0 + l16;
        if (rowA >= n) rowA = n - 1;         // clamp load, keep EXEC all-ones
        const float* xrow = x + (size_t)rowA * C_IN;

        v8f acc0 = {}, acc1 = {}, acc2 = {}, acc3 = {};

        #pragma unroll 4
        for (int k0 = 0; k0 < C_IN; k0 += 4) {
            const int kk = k0 + 2 * half;
            v2f a = *(const v2f*)(xrow + kk);

            const float* wp = lds_w + kk * C_OUT;   // rows kk and kk+1 (LDS)
            v2f b0 = { wp[l16 +  0], wp[C_OUT + l16 +  0] };
            v2f b1 = { wp[l16 + 16], wp[C_OUT + l16 + 16] };
            v2f b2 = { wp[l16 + 32], wp[C_OUT + l16 + 32] };
            v2f b3 = { wp[l16 + 48], wp[C_OUT + l16 + 48] };

            acc0 = __builtin_amdgcn_wmma_f32_16x16x4_f32(false, a, false, b0,
                                                         (short)0, acc0, false, false);
            acc1 = __builtin_amdgcn_wmma_f32_16x16x4_f32(false, a, false, b1,
                                                         (short)0, acc1, false, false);
            acc2 = __builtin_amdgcn_wmma_f32_16x16x4_f32(false, a, false, b2,
                                                         (short)0, acc2, false, false);
            acc3 = __builtin_amdgcn_wmma_f32_16x16x4_f32(false, a, false, b3,
                                                         (short)0, acc3, false, false);
        }

        // C/D layout: VGPR r holds row M = r + 8*half, column N = l16 (+tile*16)
        if (m0 + 16 <= n) {
            #pragma unroll
            for (int r = 0; r < 8; ++r) {
                float* hp = h + (size_t)(m0 + r + 8 * half) * C_OUT + l16;
                hp[0]  = acc0[r];
                hp[16] = acc1[r];
                hp[32] = acc2[r];
                hp[48] = acc3[r];
            }
        } else {
            #pragma unroll
            for (int r = 0; r < 8; ++r) {
                const int m = m0 + r + 8 * half;
                if (m < n) {
                    float* hp = h + (size_t)m * C_OUT + l16;
                    hp[0]  = acc0[r];
                    hp[16] = acc1[r];
                    hp[32] = acc2[r];
                    hp[48] = acc3[r];
                }
            }
        }
    }
}

// ---------------------------------------------------------------------------
// Kernel 2: per-node attention projections ar = h @ attw[:64], ac = h @ attw[64:]
// ---------------------------------------------------------------------------
__global__ __launch_bounds__(256) void k_nodeatt(const float* __restrict__ h,
                                                 const float* __restrict__ attw, // [128,4]
                                                 float* __restrict__ ar,
                                                 float* __restrict__ ac,
                                                 int n) {
    int i = blockIdx.x * blockDim.x + threadIdx.x;
    if (i >= n) return;
    const float* hp = h + (size_t)i * C_OUT;
    float sr0 = 0, sr1 = 0, sr2 = 0, sr3 = 0;
    float sc0 = 0, sc1 = 0, sc2 = 0, sc3 = 0;
    #pragma unroll 8
    for (int j = 0; j < C_OUT; ++j) {
        float v = hp[j];
        const float* wr = attw + j * HEADS;
        const float* wc = attw + (C_OUT + j) * HEADS;
        sr0 += v * wr[0]; sr1 += v * wr[1]; sr2 += v * wr[2]; sr3 += v * wr[3];
        sc0 += v * wc[0]; sc1 += v * wc[1]; sc2 += v * wc[2]; sc3 += v * wc[3];
    }
    float* arp = ar + (size_t)i * HEADS;
    float* acp = ac + (size_t)i * HEADS;
    arp[0] = sr0; arp[1] = sr1; arp[2] = sr2; arp[3] = sr3;
    acp[0] = sc0; acp[1] = sc1; acp[2] = sc2; acp[3] = sc3;
}

// ---------------------------------------------------------------------------
// Kernel 3: edge logits (leaky relu, slope 10 on negatives) + atomic segment max
// ---------------------------------------------------------------------------
__global__ __launch_bounds__(256) void k_edge_logits(const long long* __restrict__ ei,
                                                     const float* __restrict__ eattr,
                                                     const float* __restrict__ ar,
                                                     const float* __restrict__ ac,
                                                     float* __restrict__ alpha,
                                                     float* __restrict__ amax,
                                                     int E, int n) {
    int e = blockIdx.x * blockDim.x + threadIdx.x;
    if (e >= E + n) return;
    int r, c; float w;
    if (e < E) { r = (int)ei[e]; c = (int)ei[(size_t)E + e]; w = fabsf(eattr[e]); }
    else       { r = c = e - E; w = 1.0f; }
    const float* arp = ar + (size_t)r * HEADS;
    const float* acp = ac + (size_t)c * HEADS;
    float* ap = alpha + (size_t)e * HEADS;
    #pragma unroll
    for (int t = 0; t < HEADS; ++t) {
        float v = (arp[t] + acp[t]) * w;
        v = (v >= 0.0f) ? v : 10.0f * v;
        ap[t] = v;
        atomicMax(&amax[(size_t)r * HEADS + t], v);
    }
}

// ---------------------------------------------------------------------------
// Kernel 4: ex = exp(alpha - amax[row]); denom[row] += ex  (in-place on alpha)
// ---------------------------------------------------------------------------
__global__ __launch_bounds__(256) void k_edge_exp(const long long* __restrict__ ei,
                                                  float* __restrict__ alpha,
                                                  const float* __restrict__ amax,
                                                  float* __restrict__ denom,
                                                  int E, int n) {
    int e = blockIdx.x * blockDim.x + threadIdx.x;
    if (e >= E + n) return;
    int r = (e < E) ? (int)ei[e] : (e - E);
    float* ap = alpha + (size_t)e * HEADS;
    const float* mp = amax + (size_t)r * HEADS;
    #pragma unroll
    for (int t = 0; t < HEADS; ++t) {
        float ex = __expf(ap[t] - mp[t]);
        ap[t] = ex;
        atomicAdd(&denom[(size_t)r * HEADS + t], ex);
    }
}

// ---------------------------------------------------------------------------
// Kernel 5: head-mix a_e = sum_t (ex/denom) * softmax(att_conv_w)[t];
// also emit new_edge_index (int64) outputs.
// ---------------------------------------------------------------------------
__global__ __launch_bounds__(256) void k_edge_a(const long long* __restrict__ ei,
                                                const float* __restrict__ alpha,
                                                const float* __restrict__ denom,
                                                const float* __restrict__ acw, // [4]
                                                float* __restrict__ a_out,
                                                long long* __restrict__ nei,
                                                int E, int n) {
    int e = blockIdx.x * blockDim.x + threadIdx.x;
    int Et = E + n;
    if (e >= Et) return;
    int r, c;
    if (e < E) { r = (int)ei[e]; c = (int)ei[(size_t)E + e]; }
    else       { r = c = e - E; }

    float w0 = acw[0], w1 = acw[1], w2 = acw[2], w3 = acw[3];
    float m = fmaxf(fmaxf(w0, w1), fmaxf(w2, w3));
    float e0 = __expf(w0 - m), e1 = __expf(w1 - m), e2 = __expf(w2 - m), e3 = __expf(w3 - m);
    float inv = 1.0f / (e0 + e1 + e2 + e3);

    const float* ap = alpha + (size_t)e * HEADS;
    const float* dp = denom + (size_t)r * HEADS;
    float a = (ap[0] / dp[0]) * (e0 * inv)
            + (ap[1] / dp[1]) * (e1 * inv)
            + (ap[2] / dp[2]) * (e2 * inv)
            + (ap[3] / dp[3]) * (e3 * inv);
    a_out[e] = a;
    nei[e]              = (long long)r;
    nei[(size_t)Et + e] = (long long)c;
}

// ---------------------------------------------------------------------------
// Kernel 6: scatter out[row] += a * h[col]; 4 threads per edge, 16 floats each
// ---------------------------------------------------------------------------
__global__ __launch_bounds__(256) void k_scatter(const long long* __restrict__ ei,
                                                 const float* __restrict__ a_out,
                                                 const float* __restrict__ h,
                                                 float* __restrict__ out,
                                                 int E, int n) {
    int tid = blockIdx.x * blockDim.x + threadIdx.x;
    int Et = E + n;
    int e = tid >> 2;
    if (e >= Et) return;
    int part = tid & 3;
    int r, c;
    if (e < E) { r = (int)ei[e]; c = (int)ei[(size_t)E + e]; }
    else       { r = c = e - E; }
    float a = a_out[e];
    const float* hp = h + (size_t)c * C_OUT + part * 16;
    float* op = out + (size_t)r * C_OUT + part * 16;
    #pragma unroll
    for (int j = 0; j < 16; ++j) {
        atomicAdd(&op[j], a * hp[j]);
    }
}

// ---------------------------------------------------------------------------
extern "C" void kernel_launch(void* const* d_in, const int* in_sizes, int n_in,
                              void* d_out, int out_size, void* d_ws, size_t ws_size,
                              hipStream_t stream) {
    const float*     x     = (const float*)d_in[0];
    const long long* ei    = (const long long*)d_in[1];
    const float*     eattr = (const float*)d_in[2];
    const float*     wgt   = (const float*)d_in[3];
    const float*     attw  = (const float*)d_in[4];
    const float*     acw   = (const float*)d_in[5];
    const float*     bias  = (const float*)d_in[6];

    const int n  = in_sizes[0] / C_IN;   // 50000
    const int E  = in_sizes[2];          // 800000
    const int Et = E + n;

    // output layout (flat, in return order): out[n*64] f32, a[Et] f32, nei[2*Et] i64
    char*      ob    = (char*)d_out;
    float*     out   = (float*)ob;
    float*     a_out = (float*)(ob + (size_t)n * C_OUT * sizeof(float));
    long long* nei   = (long long*)(ob + ((size_t)n * C_OUT + (size_t)Et) * sizeof(float));

    // workspace layout (f32)
    float* ws    = (float*)d_ws;
    float* h     = ws;                               // n*64
    float* ar    = h     + (size_t)n * C_OUT;        // n*4
    float* ac    = ar    + (size_t)n * HEADS;        // n*4
    float* amax  = ac    + (size_t)n * HEADS;        // n*4
    float* denom = amax  + (size_t)n * HEADS;        // n*4
    float* alpha = denom + (size_t)n * HEADS;        // Et*4

    {
        int total = n * C_OUT;
        k_init<<<(total + 255) / 256, 256, 0, stream>>>(out, bias, amax, denom, n);
    }
    {
        int blocks = (n + 127) / 128;   // 8 waves * 16 rows per block
        k_gemm<<<blocks, 256, 0, stream>>>(x, wgt, h, n);
    }
    k_nodeatt<<<(n + 255) / 256, 256, 0, stream>>>(h, attw, ar, ac, n);
    k_edge_logits<<<(Et + 255) / 256, 256, 0, stream>>>(ei, eattr, ar, ac, alpha, amax, E, n);
    k_edge_exp<<<(Et + 255) / 256, 256, 0, stream>>>(ei, alpha, amax, denom, E, n);
    k_edge_a<<<(Et + 255) / 256, 256, 0, stream>>>(ei, alpha, denom, acw, a_out, nei, E, n);
    {
        size_t threads = (size_t)Et * 4;
        k_scatter<<<(int)((threads + 255) / 256), 256, 0, stream>>>(ei, a_out, h, out, E, n);
    }
}